// SemGraphConv_83107617178280
// MI455X (gfx1250) — compile-verified
//
#include <hip/hip_runtime.h>
#include <math.h>

namespace {

constexpr int kF  = 128;      // F_IN == F_OUT
constexpr int kNB = 512;      // partial-reduction blocks

typedef __attribute__((ext_vector_type(2))) float v2f;
typedef __attribute__((ext_vector_type(8))) float v8f;

// out[n, f] = bias[f]
__global__ void k_init_out(const float* __restrict__ bias, float* __restrict__ out, int total) {
  int i = blockIdx.x * blockDim.x + threadIdx.x;
  if (i < total) out[i] = bias[i & (kF - 1)];
}

// wsum = W0 + W1   (exploits linearity of the two update_all passes)
__global__ void k_wsum(const float* __restrict__ w, float* __restrict__ wsum) {
  int i = blockIdx.x * blockDim.x + threadIdx.x;
  if (i < kF * kF) wsum[i] = w[i] + w[kF * kF + i];
}

// ---- global softmax over all E edge scalars: max pass ----
__global__ void k_pmax(const float* __restrict__ ef, float* __restrict__ part, int e_cnt) {
  __shared__ float s[256];
  float m = -3.4e38f;
  for (int i = blockIdx.x * blockDim.x + threadIdx.x; i < e_cnt; i += gridDim.x * blockDim.x)
    m = fmaxf(m, ef[i]);
  s[threadIdx.x] = m;
  __syncthreads();
  for (int off = 128; off > 0; off >>= 1) {
    if ((int)threadIdx.x < off) s[threadIdx.x] = fmaxf(s[threadIdx.x], s[threadIdx.x + off]);
    __syncthreads();
  }
  if (threadIdx.x == 0) part[blockIdx.x] = s[0];
}

__global__ void k_fmax(const float* __restrict__ part, float* __restrict__ scal) {
  __shared__ float s[kNB];
  s[threadIdx.x] = part[threadIdx.x];
  __syncthreads();
  for (int off = kNB / 2; off > 0; off >>= 1) {
    if ((int)threadIdx.x < off) s[threadIdx.x] = fmaxf(s[threadIdx.x], s[threadIdx.x + off]);
    __syncthreads();
  }
  if (threadIdx.x == 0) scal[0] = s[0];
}

// ---- sum of exp(x - max) pass ----
__global__ void k_psum(const float* __restrict__ ef, const float* __restrict__ scal,
                       float* __restrict__ part, int e_cnt) {
  __shared__ float s[256];
  const float mx = scal[0];
  float a = 0.0f;
  for (int i = blockIdx.x * blockDim.x + threadIdx.x; i < e_cnt; i += gridDim.x * blockDim.x)
    a += __expf(ef[i] - mx);
  s[threadIdx.x] = a;
  __syncthreads();
  for (int off = 128; off > 0; off >>= 1) {
    if ((int)threadIdx.x < off) s[threadIdx.x] += s[threadIdx.x + off];
    __syncthreads();
  }
  if (threadIdx.x == 0) part[blockIdx.x] = s[0];
}

__global__ void k_fsum(const float* __restrict__ part, float* __restrict__ scal) {
  __shared__ float s[kNB];
  s[threadIdx.x] = part[threadIdx.x];
  __syncthreads();
  for (int off = kNB / 2; off > 0; off >>= 1) {
    if ((int)threadIdx.x < off) s[threadIdx.x] += s[threadIdx.x + off];
    __syncthreads();
  }
  if (threadIdx.x == 0) scal[1] = 1.0f / s[0];
}

// ---- WMMA GEMM: C[n_nodes x 128] = A[n_nodes x 128] * B[128 x 128] (f32) ----
// One wave per 16x16 output tile; 8 waves/block cover all 8 N-tiles of a 16-row stripe.
__global__ void __launch_bounds__(256) k_gemm(const float* __restrict__ A,
                                              const float* __restrict__ B,
                                              float* __restrict__ C, int n_rows) {
  const int lane = threadIdx.x & 31;
  const int wave = threadIdx.x >> 5;
  const int half = lane >> 4;     // 0: lanes 0-15, 1: lanes 16-31
  const int lo   = lane & 15;
  const int m0   = blockIdx.x * 16;
  const int n0   = wave * 16;

  // Clamp the A row so every lane issues in-bounds loads (EXEC must stay all-1s
  // through the WMMAs); duplicate rows in the tail tile are discarded at store.
  int arow_i = m0 + lo;
  if (arow_i >= n_rows) arow_i = n_rows - 1;
  const float* __restrict__ arow = A + (size_t)arow_i * kF;
  const float* __restrict__ bcol = B + n0 + lo;   // lane <-> N for B

  v8f acc = {0.f, 0.f, 0.f, 0.f, 0.f, 0.f, 0.f, 0.f};
#pragma unroll
  for (int k0 = 0; k0 < kF; k0 += 4) {
    const int k = k0 + 2 * half;          // A: lanes 16-31 hold K=2,3 of the step
    v2f a;
    a.x = arow[k];
    a.y = arow[k + 1];
    v2f b;                                 // B: VGPR r <-> K = 2*half + r
    b.x = bcol[(size_t)k * kF];
    b.y = bcol[(size_t)(k + 1) * kF];
    acc = __builtin_amdgcn_wmma_f32_16x16x4_f32(
        /*neg_a=*/false, a, /*neg_b=*/false, b,
        /*c_mod=*/(short)0, acc, /*reuse_a=*/false, /*reuse_b=*/false);
  }

  // C/D layout: VGPR r -> row (r + 8*half), lane -> column
#pragma unroll
  for (int r = 0; r < 8; ++r) {
    const int row = m0 + r + 8 * half;
    if (row < n_rows) C[(size_t)row * kF + n0 + lo] = acc[r];
  }
}

// ---- scatter: one wave per edge, lane f4-chunk per lane; atomics resolve in L2 ----
__global__ void __launch_bounds__(256) k_scatter(const float* __restrict__ hw,
                                                 const float* __restrict__ ef,
                                                 const int* __restrict__ src,
                                                 const int* __restrict__ dst,
                                                 const float* __restrict__ scal,
                                                 float* __restrict__ out, int e_cnt) {
  const int lane = threadIdx.x & 31;
  const int wave = threadIdx.x >> 5;
  const int e    = blockIdx.x * 8 + wave;
  if (e >= e_cnt) return;

  const float w = __expf(ef[e] - scal[0]) * scal[1];
  const int s = src[e];
  const int d = dst[e];

  const float4 v = *reinterpret_cast<const float4*>(hw + (size_t)s * kF + lane * 4);
  float* o = out + (size_t)d * kF + lane * 4;
  atomicAdd(o + 0, w * v.x);
  atomicAdd(o + 1, w * v.y);
  atomicAdd(o + 2, w * v.z);
  atomicAdd(o + 3, w * v.w);
}

}  // namespace

extern "C" void kernel_launch(void* const* d_in, const int* in_sizes, int n_in,
                              void* d_out, int out_size, void* d_ws, size_t ws_size,
                              hipStream_t stream) {
  const float* h    = (const float*)d_in[0];   // [N, 128]
  const float* ef   = (const float*)d_in[1];   // [E]
  const int*   src  = (const int*)d_in[2];     // [E]
  const int*   dst  = (const int*)d_in[3];     // [E]
  const float* w    = (const float*)d_in[4];   // [2, 128, 128]
  const float* bias = (const float*)d_in[5];   // [128]
  float* out = (float*)d_out;                  // [N, 128]

  const int n_nodes = in_sizes[0] / kF;
  const int e_cnt   = in_sizes[1];

  // Workspace layout (all offsets 16B-aligned where it matters):
  char* ws = (char*)d_ws;
  float* scal = (float*)ws;                       // [0]=max, [1]=1/sum
  float* part = (float*)(ws + 1024);              // kNB partials
  float* wsum = (float*)(ws + 4096);              // 128x128 = 64 KB
  float* hw   = (float*)(ws + 4096 + 65536);      // [N,128] f32 (16B-aligned)

  const int total_out = n_nodes * kF;

  // 1) out = bias (fresh every call; harness does not re-zero d_out)
  k_init_out<<<(total_out + 255) / 256, 256, 0, stream>>>(bias, out, total_out);

  // 2) Wsum = W0 + W1
  k_wsum<<<(kF * kF + 255) / 256, 256, 0, stream>>>(w, wsum);

  // 3-4) global max of edge_feat
  k_pmax<<<kNB, 256, 0, stream>>>(ef, part, e_cnt);
  k_fmax<<<1, kNB, 0, stream>>>(part, scal);

  // 5-6) sum exp(x - max), store reciprocal
  k_psum<<<kNB, 256, 0, stream>>>(ef, scal, part, e_cnt);
  k_fsum<<<1, kNB, 0, stream>>>(part, scal);

  // 7) hw = h @ Wsum via v_wmma_f32_16x16x4_f32
  k_gemm<<<(n_nodes + 15) / 16, 256, 0, stream>>>(h, wsum, hw, n_nodes);

  // 8) out[dst] += softmax(ef)[e] * hw[src]
  k_scatter<<<(e_cnt + 7) / 8, 256, 0, stream>>>(hw, ef, src, dst, scal, out, e_cnt);
}